// TreeAttention_37804302139410
// MI455X (gfx1250) — compile-verified
//
#include <hip/hip_runtime.h>
#include <hip/hip_bf16.h>
#include <math.h>

typedef __attribute__((ext_vector_type(2))) float v2f;
typedef __attribute__((ext_vector_type(8))) float v8f;

// D = A(16x4) * B(4x16) + C, all f32, full precision on the CDNA5 matrix pipe.
__device__ inline v8f wmma_f32(v2f a, v2f b, v8f c) {
  return __builtin_amdgcn_wmma_f32_16x16x4_f32(false, a, false, b, (short)0, c,
                                               false, false);
}

__device__ inline float hmax16(float v) {
  // xor masks 1..8 keep lanes 0-15 and 16-31 separate (wave32)
  for (int off = 8; off >= 1; off >>= 1) v = fmaxf(v, __shfl_xor(v, off, 32));
  return v;
}
__device__ inline float hsum16(float v) {
  for (int off = 8; off >= 1; off >>= 1) v += __shfl_xor(v, off, 32);
  return v;
}

// ---------------------------------------------------------------------------
// Kernel 1: routing + gating -> c[b*s, t*8+l] = leaf_prob * gate_weight
// One block per (b,s) row (4096 blocks, 128 threads). x row staged in LDS.
// ---------------------------------------------------------------------------
__global__ void tree_route_kernel(const float* __restrict__ x,
                                  const float* __restrict__ dw,   // [84,1024]
                                  const float* __restrict__ db,   // [84]
                                  const float* __restrict__ ntl,  // [84]
                                  const float* __restrict__ gw,   // [1024,12]
                                  const float* __restrict__ gb,   // [12]
                                  float* __restrict__ cout) {     // [4096,96]
  __shared__ float xrow[1024];
  __shared__ float dvals[84];
  __shared__ float glog[12];
  __shared__ float wts[12];
  const int r = blockIdx.x;
  const int tid = threadIdx.x;
  const float* xr = x + (size_t)r * 1024;
  for (int i = tid; i < 256; i += 128)
    ((float4*)xrow)[i] = ((const float4*)xr)[i];
  __syncthreads();

  if (tid < 84) {  // decision nodes: tid = t*7 + n
    const float* w = dw + (size_t)tid * 1024;
    float acc = 0.f;
    for (int i = 0; i < 1024; ++i) acc = fmaf(xrow[i], w[i], acc);
    float z = ntl[tid] + 0.5413f;
    float temp = log1pf(__expf(z));              // TEMPERATURE * softplus
    float d = (acc + db[tid]) / temp;
    dvals[tid] = 1.f / (1.f + __expf(-d));       // sigmoid
  } else if (tid < 96) {                         // gate logits
    int t = tid - 84;
    float acc = 0.f;
    for (int i = 0; i < 1024; ++i) acc = fmaf(xrow[i], gw[i * 12 + t], acc);
    glog[t] = acc + gb[t];
  }
  __syncthreads();

  if (tid < 12) {  // softmax over T=12
    float mx = -3.4e38f;
    for (int t = 0; t < 12; ++t) mx = fmaxf(mx, glog[t]);
    float s = 0.f;
    for (int t = 0; t < 12; ++t) s += __expf(glog[t] - mx);
    wts[tid] = __expf(glog[tid] - mx) / s;
  }
  __syncthreads();

  if (tid < 96) {  // leaf probs via root-to-leaf path product, depth 3
    int t = tid >> 3, l = tid & 7;
    int node = 0;
    float p = 1.f;
    for (int d = 0; d < 3; ++d) {
      int bit = (l >> (2 - d)) & 1;
      float pv = dvals[t * 7 + node];
      p *= bit ? (1.f - pv) : pv;   // PATH_DIR = 1-bit
      node = 2 * node + 1 + bit;
    }
    cout[(size_t)r * 96 + tid] = p * wts[t];
  }
}

// ---------------------------------------------------------------------------
// Kernel 2: qkv[h] = c[4096,96] @ leaf_flat[h][96,1024]   (f32 WMMA, K=96)
// One wave per 64x16 output block (4 M-tiles): strided B fragments loaded
// once per K-step and reused by 4 WMMAs.
// ---------------------------------------------------------------------------
__global__ void qkv_gemm_kernel(const float* __restrict__ c,
                                const float* __restrict__ leaf,  // [3,96,1024]
                                float* __restrict__ qkv) {       // [3,4096,1024]
  const int lane = threadIdx.x;
  const int half = lane >> 4;
  const int l15 = lane & 15;
  const int tm = blockIdx.x * 64;
  const int tn = blockIdx.y * 16;
  const int h = blockIdx.z;
  const float* W = leaf + (size_t)h * 96 * 1024;
  float* dst = qkv + (size_t)h * 4096 * 1024;
  const float* ar0 = c + (size_t)(tm + 0  + l15) * 96;
  const float* ar1 = c + (size_t)(tm + 16 + l15) * 96;
  const float* ar2 = c + (size_t)(tm + 32 + l15) * 96;
  const float* ar3 = c + (size_t)(tm + 48 + l15) * 96;
  v8f acc0 = {}, acc1 = {}, acc2 = {}, acc3 = {};
  for (int kk = 0; kk < 24; ++kk) {
    int k0 = kk * 4 + half * 2;
    v2f b;
    b.x = W[(size_t)k0 * 1024 + tn + l15];
    b.y = W[(size_t)(k0 + 1) * 1024 + tn + l15];
    float4 a0 = ((const float4*)ar0)[kk];
    float4 a1 = ((const float4*)ar1)[kk];
    float4 a2 = ((const float4*)ar2)[kk];
    float4 a3 = ((const float4*)ar3)[kk];
    v2f f0, f1, f2, f3;
    f0.x = half ? a0.z : a0.x; f0.y = half ? a0.w : a0.y;
    f1.x = half ? a1.z : a1.x; f1.y = half ? a1.w : a1.y;
    f2.x = half ? a2.z : a2.x; f2.y = half ? a2.w : a2.y;
    f3.x = half ? a3.z : a3.x; f3.y = half ? a3.w : a3.y;
    acc0 = wmma_f32(f0, b, acc0);
    acc1 = wmma_f32(f1, b, acc1);
    acc2 = wmma_f32(f2, b, acc2);
    acc3 = wmma_f32(f3, b, acc3);
  }
  for (int v = 0; v < 8; ++v) {
    int row = tm + v + half * 8;
    dst[(size_t)(row +  0) * 1024 + tn + l15] = acc0[v];
    dst[(size_t)(row + 16) * 1024 + tn + l15] = acc1[v];
    dst[(size_t)(row + 32) * 1024 + tn + l15] = acc2[v];
    dst[(size_t)(row + 48) * 1024 + tn + l15] = acc3[v];
  }
}

// ---------------------------------------------------------------------------
// Kernel 3: causal flash attention, one wave per (b, h, 16-query tile).
// S-tile via 16 f32 WMMA (dh=64, K=4 per step); online softmax; P bounced
// through LDS to re-layout C/D form -> A-fragment form for P*V.
// ---------------------------------------------------------------------------
__global__ void attn_kernel(const float* __restrict__ qkv,  // [3,4096,1024]
                            float* __restrict__ ctx) {       // [4096,1024]
  __shared__ float ldsP[256];
  const int lane = threadIdx.x;
  const int half = lane >> 4;
  const int l15 = lane & 15;
  const int qm = blockIdx.x * 16;
  const int h = blockIdx.y;
  const int bb = blockIdx.z;
  const float* qmat = qkv;
  const float* kmat = qkv + (size_t)4096 * 1024;
  const float* vmat = qkv + (size_t)2 * 4096 * 1024;

  // Preload 16 A-fragments of Q (this wave's 16 query rows, dh=64)
  const float* qrow =
      qmat + (size_t)(bb * 1024 + qm + l15) * 1024 + h * 64;
  v2f qa[16];
  for (int kk = 0; kk < 16; ++kk) {
    float4 t = ((const float4*)qrow)[kk];
    qa[kk].x = half ? t.z : t.x;
    qa[kk].y = half ? t.w : t.y;
  }

  v8f acc0 = {}, acc1 = {}, acc2 = {}, acc3 = {};
  float m[8], l[8];
  for (int v = 0; v < 8; ++v) { m[v] = -3.0e38f; l[v] = 0.f; }

  const int ntile = qm / 16 + 1;  // causal: only key tiles <= query tile
  for (int kt = 0; kt < ntile; ++kt) {
    const int kn0 = kt * 16;
    // ---- S = (Q K^T) * scale ----
    const float* krow =
        kmat + (size_t)(bb * 1024 + kn0 + l15) * 1024 + h * 64;
    v8f s = {};
    for (int kk = 0; kk < 16; ++kk) {
      float4 t = ((const float4*)krow)[kk];
      v2f bf; bf.x = half ? t.z : t.x; bf.y = half ? t.w : t.y;
      s = wmma_f32(qa[kk], bf, s);
    }
    // ---- online softmax (rows live in the C/D lane layout) ----
    float pv[8];
    for (int v = 0; v < 8; ++v) {
      float sv = s[v] * 0.125f;              // 1/sqrt(64)
      int row = qm + v + half * 8;
      int col = kn0 + l15;
      if (col > row) sv = -3.0e38f;          // causal mask
      float cmax = hmax16(sv);
      float nm = fmaxf(m[v], cmax);
      float alpha = __expf(m[v] - nm);
      float p = __expf(sv - nm);
      float ps = hsum16(p);
      l[v] = l[v] * alpha + ps;
      m[v] = nm;
      acc0[v] *= alpha; acc1[v] *= alpha; acc2[v] *= alpha; acc3[v] *= alpha;
      pv[v] = p;
    }
    __syncthreads();
    for (int v = 0; v < 8; ++v) ldsP[(v + half * 8) * 16 + l15] = pv[v];
    __syncthreads();
    // ---- ctx += P * V  (K=16 keys -> 4 WMMA steps, 4 d-tiles of 16) ----
    for (int kk = 0; kk < 4; ++kk) {
      v2f a;
      a.x = ldsP[l15 * 16 + kk * 4 + half * 2];
      a.y = ldsP[l15 * 16 + kk * 4 + half * 2 + 1];
      const int k0 = kn0 + kk * 4 + half * 2;
      const float* v0 =
          vmat + (size_t)(bb * 1024 + k0) * 1024 + h * 64 + l15;
      const float* v1 = v0 + 1024;
      v2f b0; b0.x = v0[0];  b0.y = v1[0];
      v2f b1; b1.x = v0[16]; b1.y = v1[16];
      v2f b2; b2.x = v0[32]; b2.y = v1[32];
      v2f b3; b3.x = v0[48]; b3.y = v1[48];
      acc0 = wmma_f32(a, b0, acc0);
      acc1 = wmma_f32(a, b1, acc1);
      acc2 = wmma_f32(a, b2, acc2);
      acc3 = wmma_f32(a, b3, acc3);
    }
    __syncthreads();
  }

  for (int v = 0; v < 8; ++v) {
    float inv = 1.f / l[v];
    int row = qm + v + half * 8;
    float* crow = ctx + (size_t)(bb * 1024 + row) * 1024 + h * 64 + l15;
    crow[0]  = acc0[v] * inv;
    crow[16] = acc1[v] * inv;
    crow[32] = acc2[v] * inv;
    crow[48] = acc3[v] * inv;
  }
}

// ---------------------------------------------------------------------------
// Kernel 4: out = ctx[4096,1024] @ out_w[1024,1024] + out_b (f32 WMMA, K=1024)
// One wave per 64x16 output block (4 M-tiles): strided B fragments loaded
// once per K-step, reused by 4 WMMAs against contiguous float4 A fragments.
// ---------------------------------------------------------------------------
__global__ void outproj_kernel(const float* __restrict__ ctx,
                               const float* __restrict__ W,
                               const float* __restrict__ bias,
                               float* __restrict__ out) {
  const int lane = threadIdx.x;
  const int half = lane >> 4;
  const int l15 = lane & 15;
  const int tm = blockIdx.x * 64;
  const int tn = blockIdx.y * 16;
  const float* ar0 = ctx + (size_t)(tm + 0  + l15) * 1024;
  const float* ar1 = ctx + (size_t)(tm + 16 + l15) * 1024;
  const float* ar2 = ctx + (size_t)(tm + 32 + l15) * 1024;
  const float* ar3 = ctx + (size_t)(tm + 48 + l15) * 1024;
  v8f acc0 = {}, acc1 = {}, acc2 = {}, acc3 = {};
  for (int kk = 0; kk < 256; ++kk) {
    int k0 = kk * 4 + half * 2;
    v2f b;
    b.x = W[(size_t)k0 * 1024 + tn + l15];
    b.y = W[(size_t)(k0 + 1) * 1024 + tn + l15];
    float4 a0 = ((const float4*)ar0)[kk];
    float4 a1 = ((const float4*)ar1)[kk];
    float4 a2 = ((const float4*)ar2)[kk];
    float4 a3 = ((const float4*)ar3)[kk];
    v2f f0, f1, f2, f3;
    f0.x = half ? a0.z : a0.x; f0.y = half ? a0.w : a0.y;
    f1.x = half ? a1.z : a1.x; f1.y = half ? a1.w : a1.y;
    f2.x = half ? a2.z : a2.x; f2.y = half ? a2.w : a2.y;
    f3.x = half ? a3.z : a3.x; f3.y = half ? a3.w : a3.y;
    acc0 = wmma_f32(f0, b, acc0);
    acc1 = wmma_f32(f1, b, acc1);
    acc2 = wmma_f32(f2, b, acc2);
    acc3 = wmma_f32(f3, b, acc3);
  }
  const float bv = bias[tn + l15];
  for (int v = 0; v < 8; ++v) {
    int row = tm + v + half * 8;
    out[(size_t)(row +  0) * 1024 + tn + l15] = acc0[v] + bv;
    out[(size_t)(row + 16) * 1024 + tn + l15] = acc1[v] + bv;
    out[(size_t)(row + 32) * 1024 + tn + l15] = acc2[v] + bv;
    out[(size_t)(row + 48) * 1024 + tn + l15] = acc3[v] + bv;
  }
}

// ---------------------------------------------------------------------------
extern "C" void kernel_launch(void* const* d_in, const int* in_sizes, int n_in,
                              void* d_out, int out_size, void* d_ws,
                              size_t ws_size, hipStream_t stream) {
  const float* x    = (const float*)d_in[0];
  const float* dw   = (const float*)d_in[1];
  const float* db   = (const float*)d_in[2];
  const float* ntl  = (const float*)d_in[3];
  const float* gw   = (const float*)d_in[4];
  const float* gb   = (const float*)d_in[5];
  const float* leaf = (const float*)d_in[6];
  const float* ow   = (const float*)d_in[7];
  const float* ob   = (const float*)d_in[8];
  float* out = (float*)d_out;

  float* ws   = (float*)d_ws;
  float* cbuf = ws;                                  // 4096*96
  float* qkv  = ws + 4096 * 96;                      // 3*4096*1024
  float* ctx  = qkv + (size_t)3 * 4096 * 1024;       // 4096*1024

  tree_route_kernel<<<4096, 128, 0, stream>>>(x, dw, db, ntl, gw, gb, cbuf);
  qkv_gemm_kernel<<<dim3(64, 64, 3), 32, 0, stream>>>(cbuf, leaf, qkv);
  attn_kernel<<<dim3(64, 16, 4), 32, 0, stream>>>(qkv, ctx);
  outproj_kernel<<<dim3(64, 64), 32, 0, stream>>>(ctx, ow, ob, out);
}